// AttentionDecoderLSTMCell_6081673691720
// MI455X (gfx1250) — compile-verified
//
#include <hip/hip_runtime.h>
#include <hip/hip_bf16.h>
#include <cmath>

#define BB 64
#define LL 4096
#define DD 512
#define EE 256
#define VV 32000
#define KXX 768   // D + E

typedef __attribute__((ext_vector_type(2))) float v2f;
typedef __attribute__((ext_vector_type(8))) float v8f;

// ---------------------------------------------------------------------------
// WMMA helper: D = A(16x4, f32) * B(4x16, f32) + C(16x16, f32)
// A layout (32-bit 16x4): lane%16 = M, lane/16 selects K-half, v[0..1] = K pair
// B layout (4x16):        lane%16 = N, lane/16 selects K-half
// C/D layout:             VGPR r: lanes0-15 -> M=r, lanes16-31 -> M=r+8
// ---------------------------------------------------------------------------
__device__ __forceinline__ v8f wmma4(v2f a, v2f b, v8f c) {
  return __builtin_amdgcn_wmma_f32_16x16x4_f32(false, a, false, b, (short)0, c,
                                               false, false);
}

// One wave computes a full 64-row column stripe (4 M-tiles) of A[64xK]*B[KxN].
// B register pair is reused across all 4 WMMAs -> each weight byte loaded once.
__device__ __forceinline__ void wave_col_gemm(const float* __restrict__ A, int lda,
                                              const float* __restrict__ Bm, int ldb,
                                              int K, int ncol, int khalf, int mlane,
                                              v8f acc[4]) {
  for (int k = 0; k < K; k += 4) {
    v2f b;
    b.x = Bm[(size_t)(k + khalf) * ldb + ncol];
    b.y = Bm[(size_t)(k + khalf + 1) * ldb + ncol];
#pragma unroll
    for (int t = 0; t < 4; ++t) {
      const float* Ar = A + (size_t)(t * 16 + mlane) * lda + (k + khalf);
      v2f a; a.x = Ar[0]; a.y = Ar[1];
      acc[t] = wmma4(a, b, acc[t]);
    }
  }
}

__device__ __forceinline__ float sigmoid_(float v) {
  return 1.0f / (1.0f + __expf(-v));
}

// ---------------------------------------------------------------------------
// K1: per-batch argmax over y_prev[V], then gather x = [embedding[tok] | out_prev]
// ---------------------------------------------------------------------------
__global__ __launch_bounds__(256) void k1_argmax_embed(
    const float* __restrict__ y_prev, const float* __restrict__ embedding,
    const float* __restrict__ out_prev, float* __restrict__ x) {
  int b = blockIdx.x;
  int tid = threadIdx.x;
  const float* yp = y_prev + (size_t)b * VV;

  float best = -3.4e38f;
  int bidx = VV;
  for (int v = tid; v < VV; v += 256) {
    float val = yp[v];
    if (val > best || (val == best && v < bidx)) { best = val; bidx = v; }
  }
  __shared__ float smax[256];
  __shared__ int   sidx[256];
  smax[tid] = best; sidx[tid] = bidx;
  __syncthreads();
  for (int s = 128; s > 0; s >>= 1) {
    if (tid < s) {
      float ov = smax[tid + s]; int oi = sidx[tid + s];
      if (ov > smax[tid] || (ov == smax[tid] && oi < sidx[tid])) {
        smax[tid] = ov; sidx[tid] = oi;
      }
    }
    __syncthreads();
  }
  int tok = sidx[0];
  for (int j = tid; j < EE; j += 256)
    x[(size_t)b * KXX + j] = embedding[(size_t)tok * EE + j];
  for (int j = tid; j < DD; j += 256)
    x[(size_t)b * KXX + EE + j] = out_prev[(size_t)b * DD + j];
}

// ---------------------------------------------------------------------------
// K2: fused gate GEMMs + LSTM cell.  128 waves = 4 (M) x 32 (N) tiles.
// One A (x) register pair feeds all four gate WMMAs per K-step.
// ---------------------------------------------------------------------------
__global__ __launch_bounds__(128) void k2_gates(
    const float* __restrict__ x,
    const float* __restrict__ Wf, const float* __restrict__ Wi,
    const float* __restrict__ Wg, const float* __restrict__ Wo,
    const float* __restrict__ bf, const float* __restrict__ bi,
    const float* __restrict__ bg, const float* __restrict__ bo,
    const float* __restrict__ c_prev,
    float* __restrict__ h_out, float* __restrict__ c_out) {
  int wave = (blockIdx.x * 128 + threadIdx.x) >> 5;
  int lane = threadIdx.x & 31;
  int tm = wave >> 5;        // 0..3
  int tn = wave & 31;        // 0..31
  int mrow = tm * 16 + (lane & 15);
  int ncol = tn * 16 + (lane & 15);
  int khalf = (lane >> 4) << 1;

  v8f aF = {}, aI = {}, aG = {}, aO = {};
  for (int k = 0; k < KXX; k += 4) {
    const float* Ar = x + (size_t)mrow * KXX + (k + khalf);
    v2f a; a.x = Ar[0]; a.y = Ar[1];
    size_t bo0 = (size_t)(k + khalf) * DD + ncol;
    size_t bo1 = (size_t)(k + khalf + 1) * DD + ncol;
    v2f bF; bF.x = Wf[bo0]; bF.y = Wf[bo1];
    v2f bI; bI.x = Wi[bo0]; bI.y = Wi[bo1];
    v2f bG; bG.x = Wg[bo0]; bG.y = Wg[bo1];
    v2f bO; bO.x = Wo[bo0]; bO.y = Wo[bo1];
    aF = wmma4(a, bF, aF);
    aI = wmma4(a, bI, aI);
    aG = wmma4(a, bG, aG);
    aO = wmma4(a, bO, aO);
  }

  float vbf = bf[ncol], vbi = bi[ncol], vbg = bg[ncol], vbo = bo[ncol];
#pragma unroll
  for (int r = 0; r < 8; ++r) {
    int row = tm * 16 + ((lane >> 4) << 3) + r;
    float f = sigmoid_(aF[r] + vbf);
    float i = sigmoid_(aI[r] + vbi);
    float g = tanhf(aG[r] + vbg);
    float o = sigmoid_(aO[r] + vbo);
    float c = c_prev[(size_t)row * DD + ncol] * f + i * g;
    float h = tanhf(c) * o;
    c_out[(size_t)row * DD + ncol] = c;
    h_out[(size_t)row * DD + ncol] = h;
  }
}

// ---------------------------------------------------------------------------
// K3: bvec = h @ W_e  (64x512 @ 512x512).  32 waves; each owns a full
// 64-row column stripe so W_e is read exactly once.
// ---------------------------------------------------------------------------
__global__ __launch_bounds__(128) void k3_bvec(
    const float* __restrict__ h, const float* __restrict__ We,
    float* __restrict__ bvec) {
  int wave = (blockIdx.x * 128 + threadIdx.x) >> 5;   // 0..31 = N tile
  int lane = threadIdx.x & 31;
  int mlane = lane & 15;
  int ncol = wave * 16 + mlane;
  int khalf = (lane >> 4) << 1;
  v8f acc[4] = {{}, {}, {}, {}};
  wave_col_gemm(h, DD, We, DD, DD, ncol, khalf, mlane, acc);
#pragma unroll
  for (int t = 0; t < 4; ++t)
#pragma unroll
    for (int r = 0; r < 8; ++r) {
      int row = t * 16 + ((lane >> 4) << 3) + r;
      bvec[(size_t)row * DD + ncol] = acc[t][r];
    }
}

// ---------------------------------------------------------------------------
// K4: flash-style attention: single pass over featureGrid[b] (512 MB total).
// One workgroup (8 waves) per batch.  Online softmax with per-wave running
// (m, s) and a 512-wide context accumulator held in 16 VGPRs per lane.
// ---------------------------------------------------------------------------
__global__ __launch_bounds__(256) void k4_attention(
    const float* __restrict__ fg, const float* __restrict__ bvec,
    float* __restrict__ zout) {
  int b = blockIdx.x;
  int tid = threadIdx.x;
  int lane = tid & 31;
  int wv = tid >> 5;   // 0..7

  __shared__ float s_b[DD];
  __shared__ float s_m[8], s_s[8];
  __shared__ float s_z[8][DD];

  for (int j = tid; j < DD; j += 256) s_b[j] = bvec[(size_t)b * DD + j];
  __syncthreads();

  float bv[16];
#pragma unroll
  for (int j = 0; j < 16; ++j) bv[j] = s_b[lane + 32 * j];

  const float* fgb = fg + (size_t)b * LL * DD;
  float m = -3.4e38f, s = 0.0f;
  float zacc[16];
#pragma unroll
  for (int j = 0; j < 16; ++j) zacc[j] = 0.0f;

  for (int l = wv; l < LL; l += 8) {
    const float* row = fgb + (size_t)l * DD;
    if (l + 8 < LL) __builtin_prefetch(row + 8 * DD + lane, 0, 0);
    float fv[16];
    float part = 0.0f;
#pragma unroll
    for (int j = 0; j < 16; ++j) {
      fv[j] = row[lane + 32 * j];
      part += fv[j] * bv[j];
    }
#pragma unroll
    for (int off = 16; off > 0; off >>= 1)
      part += __shfl_xor(part, off, 32);
    float e = part;
    float nm = fmaxf(m, e);
    float scale = __expf(m - nm);
    float p = __expf(e - nm);
    s = s * scale + p;
#pragma unroll
    for (int j = 0; j < 16; ++j) zacc[j] = zacc[j] * scale + p * fv[j];
    m = nm;
  }

  if (lane == 0) { s_m[wv] = m; s_s[wv] = s; }
#pragma unroll
  for (int j = 0; j < 16; ++j) s_z[wv][lane + 32 * j] = zacc[j];
  __syncthreads();

  float M = s_m[0];
#pragma unroll
  for (int w = 1; w < 8; ++w) M = fmaxf(M, s_m[w]);
  float S = 0.0f;
#pragma unroll
  for (int w = 0; w < 8; ++w) S += s_s[w] * __expf(s_m[w] - M);
  float invS = 1.0f / S;

  for (int d = tid; d < DD; d += 256) {
    float acc = 0.0f;
#pragma unroll
    for (int w = 0; w < 8; ++w) acc += s_z[w][d] * __expf(s_m[w] - M);
    zout[(size_t)b * DD + d] = acc * invS;
  }
}

// ---------------------------------------------------------------------------
// K5: out = tanh([h | z] @ W_out)  (64x1024 @ 1024x512).  32 waves, each owns
// a full 64-row column stripe; W_out read exactly once.
// ---------------------------------------------------------------------------
__global__ __launch_bounds__(128) void k5_out(
    const float* __restrict__ h, const float* __restrict__ z,
    const float* __restrict__ Wout, float* __restrict__ outv) {
  int wave = (blockIdx.x * 128 + threadIdx.x) >> 5;   // 0..31 = N tile
  int lane = threadIdx.x & 31;
  int mlane = lane & 15;
  int ncol = wave * 16 + mlane;
  int khalf = (lane >> 4) << 1;

  v8f acc[4] = {{}, {}, {}, {}};
  for (int k = 0; k < 2 * DD; k += 4) {
    int ka = k + khalf;
    const float* Abase = (ka < DD) ? h : z;
    int kk = (ka < DD) ? ka : (ka - DD);
    v2f b;
    b.x = Wout[(size_t)ka * DD + ncol];
    b.y = Wout[(size_t)(ka + 1) * DD + ncol];
#pragma unroll
    for (int t = 0; t < 4; ++t) {
      const float* Ar = Abase + (size_t)(t * 16 + mlane) * DD + kk;
      v2f a; a.x = Ar[0]; a.y = Ar[1];
      acc[t] = wmma4(a, b, acc[t]);
    }
  }
#pragma unroll
  for (int t = 0; t < 4; ++t)
#pragma unroll
    for (int r = 0; r < 8; ++r) {
      int row = t * 16 + ((lane >> 4) << 3) + r;
      outv[(size_t)row * DD + ncol] = tanhf(acc[t][r]);
    }
}

// ---------------------------------------------------------------------------
// K6: logits = out @ W_y  (64x512 @ 512x32000).  2000 waves, one per 16-wide
// N stripe covering all of M=64 -> W_y streamed exactly once.
// ---------------------------------------------------------------------------
__global__ __launch_bounds__(256) void k6_logits(
    const float* __restrict__ outv, const float* __restrict__ Wy,
    float* __restrict__ y) {
  int wave = (blockIdx.x * 256 + threadIdx.x) >> 5;   // 0..1999 = N tile
  int lane = threadIdx.x & 31;
  int mlane = lane & 15;
  int ncol = wave * 16 + mlane;
  int khalf = (lane >> 4) << 1;
  v8f acc[4] = {{}, {}, {}, {}};
  wave_col_gemm(outv, DD, Wy, VV, DD, ncol, khalf, mlane, acc);
#pragma unroll
  for (int t = 0; t < 4; ++t)
#pragma unroll
    for (int r = 0; r < 8; ++r) {
      int row = t * 16 + ((lane >> 4) << 3) + r;
      y[(size_t)row * VV + ncol] = acc[t][r];
    }
}

// ---------------------------------------------------------------------------
// K7: in-place softmax over V per batch row.
// ---------------------------------------------------------------------------
__global__ __launch_bounds__(256) void k7_softmax_v(float* __restrict__ y) {
  int b = blockIdx.x;
  int tid = threadIdx.x;
  float* yb = y + (size_t)b * VV;
  __shared__ float red[256];

  float m = -3.4e38f;
  for (int v = tid; v < VV; v += 256) m = fmaxf(m, yb[v]);
  red[tid] = m;
  __syncthreads();
  for (int s = 128; s > 0; s >>= 1) {
    if (tid < s) red[tid] = fmaxf(red[tid], red[tid + s]);
    __syncthreads();
  }
  m = red[0];
  __syncthreads();

  float sum = 0.0f;
  for (int v = tid; v < VV; v += 256) sum += __expf(yb[v] - m);
  red[tid] = sum;
  __syncthreads();
  for (int s = 128; s > 0; s >>= 1) {
    if (tid < s) red[tid] += red[tid + s];
    __syncthreads();
  }
  float inv = 1.0f / red[0];
  for (int v = tid; v < VV; v += 256) yb[v] = __expf(yb[v] - m) * inv;
}

// ---------------------------------------------------------------------------
extern "C" void kernel_launch(void* const* d_in, const int* in_sizes, int n_in,
                              void* d_out, int out_size, void* d_ws, size_t ws_size,
                              hipStream_t stream) {
  const float* featureGrid = (const float*)d_in[0];
  const float* y_prev      = (const float*)d_in[1];
  const float* out_prev    = (const float*)d_in[2];
  // d_in[3] = h_prev (unused by the reference)
  const float* c_prev      = (const float*)d_in[4];
  const float* embedding   = (const float*)d_in[5];
  const float* W_f         = (const float*)d_in[6];
  const float* W_i         = (const float*)d_in[7];
  const float* W_g         = (const float*)d_in[8];
  const float* W_o         = (const float*)d_in[9];
  const float* b_f         = (const float*)d_in[10];
  const float* b_i         = (const float*)d_in[11];
  const float* b_g         = (const float*)d_in[12];
  const float* b_o         = (const float*)d_in[13];
  const float* W_e         = (const float*)d_in[14];
  const float* W_out       = (const float*)d_in[15];
  const float* W_y         = (const float*)d_in[16];

  float* y    = (float*)d_out;                 // [64][32000]
  float* outv = y + (size_t)BB * VV;           // [64][512]
  float* h    = outv + (size_t)BB * DD;        // [64][512]
  float* c    = h + (size_t)BB * DD;           // [64][512]

  float* ws   = (float*)d_ws;
  float* x    = ws;                            // [64][768]
  float* bvec = x + (size_t)BB * KXX;          // [64][512]
  float* z    = bvec + (size_t)BB * DD;        // [64][512]

  k1_argmax_embed<<<BB, 256, 0, stream>>>(y_prev, embedding, out_prev, x);
  k2_gates<<<32, 128, 0, stream>>>(x, W_f, W_i, W_g, W_o, b_f, b_i, b_g, b_o,
                                   c_prev, h, c);
  k3_bvec<<<8, 128, 0, stream>>>(h, W_e, bvec);
  k4_attention<<<BB, 256, 0, stream>>>(featureGrid, bvec, z);
  k5_out<<<8, 128, 0, stream>>>(h, z, W_out, outv);
  k6_logits<<<250, 256, 0, stream>>>(outv, W_y, y);
  k7_softmax_v<<<BB, 256, 0, stream>>>(y);
}